// HGSSModel_29927332119165
// MI455X (gfx1250) — compile-verified
//
#include <hip/hip_runtime.h>

#define D 64
#define EPSF 1e-7f
#define MIN_NORMF 1e-15f
#define BATCH 8   // async-gather slots per 16-lane row group (8 x 256B in flight)

#if defined(__AMDGCN__) && defined(__gfx1250__)
#define USE_ASYNC_LDS 1
#endif

typedef float v2f __attribute__((ext_vector_type(2)));

// ---------------------------------------------------------------------------
// Kernel 1: x_tangent = logmap0(proj(weight), c=1)
// One wave32 per node; lane l owns elements (2l, 2l+1). Weight is streamed
// once -> non-temporal load (TH=NT) so it doesn't displace the L2-resident
// feature buffers.
// ---------------------------------------------------------------------------
__global__ void logmap_kernel(const float* __restrict__ w,
                              float* __restrict__ out, int N) {
  int wave = (blockIdx.x * blockDim.x + threadIdx.x) >> 5;
  int lane = threadIdx.x & 31;
  if (wave >= N) return;

  v2f v = __builtin_nontemporal_load(((const v2f*)(w + (size_t)wave * D)) + lane);
  float a = (lane == 0) ? 0.0f : v.x;   // element 0 excluded from y
  float b = v.y;
  float s = a * a + b * b;
  #pragma unroll
  for (int m = 16; m >= 1; m >>= 1) s += __shfl_xor(s, m, 32);

  float ysq   = s;
  float x0    = sqrtf(fmaxf(1.0f + ysq, EPSF));            // proj
  float theta = fmaxf(x0, 1.0f + EPSF);                    // logmap0 clamp
  float ach   = logf(theta + sqrtf(theta * theta - 1.0f)); // arccosh
  float ynorm = fmaxf(sqrtf(ysq), MIN_NORMF);
  float scale = ach / ynorm;

  float2 o;
  o.x = (lane == 0) ? 0.0f : scale * v.x;
  o.y = scale * v.y;
  ((float2*)(out + (size_t)wave * D))[lane] = o;
}

// ---------------------------------------------------------------------------
// Zero acc (n4 float4) + counts/cursor (N u32) in one pass. Run every call:
// the harness does not re-zero buffers between replays.
// ---------------------------------------------------------------------------
__global__ void init_kernel(float4* __restrict__ acc, unsigned* __restrict__ counts,
                            unsigned* __restrict__ cursor, int n4, int N) {
  int i = blockIdx.x * blockDim.x + threadIdx.x;
  if (i < n4) acc[i] = make_float4(0.0f, 0.0f, 0.0f, 0.0f);
  if (i < N) { counts[i] = 0u; cursor[i] = 0u; }
}

// ---------------------------------------------------------------------------
// CSR build step 1: per-row edge counts (u32 atomics, E ops total).
// ---------------------------------------------------------------------------
__global__ void count_kernel(const int* __restrict__ rows,
                             unsigned* __restrict__ counts, int E) {
  int e = blockIdx.x * blockDim.x + threadIdx.x;
  if (e < E) {
    int r = __builtin_nontemporal_load(rows + e);
    atomicAdd(&counts[r], 1u);
  }
}

// ---------------------------------------------------------------------------
// CSR build step 2a: block-local exclusive scan of counts (256/block).
// Wave32 shfl_up scan + LDS combine across the 8 waves.
// ---------------------------------------------------------------------------
__global__ void scan_block_kernel(const unsigned* __restrict__ counts,
                                  unsigned* __restrict__ rowStart,
                                  unsigned* __restrict__ blockSums, int N) {
  __shared__ unsigned wsum[8];
  int i    = blockIdx.x * 256 + threadIdx.x;
  int lane = threadIdx.x & 31;
  int wid  = threadIdx.x >> 5;

  unsigned c = (i < N) ? counts[i] : 0u;
  unsigned s = c;
  #pragma unroll
  for (int d = 1; d < 32; d <<= 1) {
    unsigned t = __shfl_up(s, d, 32);
    if (lane >= d) s += t;
  }
  if (lane == 31) wsum[wid] = s;
  __syncthreads();
  if (threadIdx.x == 0) {
    unsigned run = 0;
    #pragma unroll
    for (int k = 0; k < 8; ++k) { unsigned t = wsum[k]; wsum[k] = run; run += t; }
    blockSums[blockIdx.x] = run;
  }
  __syncthreads();
  if (i < N) rowStart[i] = s - c + wsum[wid];
}

// ---------------------------------------------------------------------------
// CSR build step 2b: serial exclusive scan of ~782 block sums (negligible).
// ---------------------------------------------------------------------------
__global__ void scan_top_kernel(unsigned* __restrict__ blockSums, int nb) {
  if (blockIdx.x == 0 && threadIdx.x == 0) {
    unsigned run = 0;
    for (int b = 0; b < nb; ++b) { unsigned t = blockSums[b]; blockSums[b] = run; run += t; }
  }
}

// ---------------------------------------------------------------------------
// CSR build step 3: scatter (col,val) into CSR order. One u32 atomic/edge.
// ---------------------------------------------------------------------------
__global__ void scatter_kernel(const float* __restrict__ vals,
                               const int*   __restrict__ rows,
                               const int*   __restrict__ cols,
                               const unsigned* __restrict__ rowStart,
                               const unsigned* __restrict__ blockSums,
                               unsigned* __restrict__ cursor,
                               int* __restrict__ cidx, float* __restrict__ cval,
                               int E) {
  int e = blockIdx.x * blockDim.x + threadIdx.x;
  if (e >= E) return;
  int r = __builtin_nontemporal_load(rows + e);
  unsigned base = rowStart[r] + blockSums[r >> 8];
  unsigned pos  = base + atomicAdd(&cursor[r], 1u);
  cidx[pos] = __builtin_nontemporal_load(cols + e);
  cval[pos] = __builtin_nontemporal_load(vals + e);
}

// ---------------------------------------------------------------------------
// CSR SpMM (gather form, ZERO float atomics): one 16-lane group per row,
// float4 per lane, register accumulation, single store per row, acc fused.
//
// gfx1250 path: batch the row gathers through GLOBAL_LOAD_ASYNC_TO_LDS_B128
// (ASYNCcnt-tracked). Each group issues up to BATCH independent 256B row
// fetches into its private LDS slots (no dest VGPRs held), drains them with
// one s_wait_asynccnt 0, then consumes at ds_load_b128 rate. This replaces a
// 1-deep dependent prefetch chain with 8 gathers in flight per group
// (128 per block). Both src and dst are L2-resident (51.2 MB vs 192 MB L2).
// ---------------------------------------------------------------------------
__global__ void csr_spmm_kernel(const int* __restrict__ cidx,
                                const float* __restrict__ cval,
                                const unsigned* __restrict__ counts,
                                const unsigned* __restrict__ rowStart,
                                const unsigned* __restrict__ blockSums,
                                const float* __restrict__ src,
                                float* __restrict__ dst,
                                float* __restrict__ acc, int N) {
  int r    = blockIdx.x * 16 + (threadIdx.x >> 4);  // 256 thr = 16 row-groups
  int lane = threadIdx.x & 15;
#if USE_ASYNC_LDS
  __shared__ __align__(16) float smem[16 * BATCH * D];   // 32 KB per block
#endif
  if (r >= N) return;

  unsigned len  = counts[r];
  unsigned base = rowStart[r] + blockSums[r >> 8];
  float4 sum = make_float4(0.0f, 0.0f, 0.0f, 0.0f);

#if USE_ASYNC_LDS
  // This group's LDS slots; low 32 bits of a flat LDS pointer are the
  // wave-relative LDS byte address (ISA 10.2 aperture rules).
  float* slot = smem + (size_t)(threadIdx.x >> 4) * (BATCH * D) + lane * 4;
  for (unsigned jb = 0; jb < len; jb += BATCH) {
    unsigned cnt = min((unsigned)BATCH, len - jb);
    for (unsigned k = 0; k < cnt; ++k) {
      int col = cidx[base + jb + k];
      unsigned long long g = (unsigned long long)(const void*)(src + (size_t)col * D + lane * 4);
      unsigned           l = (unsigned)(size_t)(slot + k * D);
      asm volatile("global_load_async_to_lds_b128 %0, %1, off"
                   :: "v"(l), "v"(g) : "memory");
    }
    asm volatile("s_wait_asynccnt 0x0" ::: "memory");
    for (unsigned k = 0; k < cnt; ++k) {
      float v = cval[base + jb + k];
      float4 x = *(const float4*)(slot + k * D);
      sum.x += v * x.x; sum.y += v * x.y; sum.z += v * x.z; sum.w += v * x.w;
    }
  }
#else
  int colNext = (len > 0) ? cidx[base] : 0;
  for (unsigned j = 0; j < len; ++j) {
    int   col = colNext;
    float v   = cval[base + j];
    if (j + 1 < len) {
      colNext = cidx[base + j + 1];
      __builtin_prefetch(src + (size_t)colNext * D + lane * 4, 0, 3);
    }
    float4 x = ((const float4*)(src + (size_t)col * D))[lane];
    sum.x += v * x.x; sum.y += v * x.y; sum.z += v * x.z; sum.w += v * x.w;
  }
#endif

  ((float4*)(dst + (size_t)r * D))[lane] = sum;          // next layer's input
  float4* ap = (float4*)(acc + (size_t)r * D);           // fused accumulator
  float4 a = ap[lane];
  a.x += sum.x; a.y += sum.y; a.z += sum.z; a.w += sum.w;
  ap[lane] = a;
}

// ---------------------------------------------------------------------------
// Fallback COO scatter SpMM (used only if ws too small for CSR arrays).
// ---------------------------------------------------------------------------
__global__ void coo_spmm_kernel(const float* __restrict__ vals,
                                const int*   __restrict__ rows,
                                const int*   __restrict__ cols,
                                const float* __restrict__ src,
                                float*       __restrict__ dst,
                                int E, int stride) {
  int group = blockIdx.x * 16 + (threadIdx.x >> 4);
  int lane  = threadIdx.x & 15;
  for (int e = group; e < E; e += stride) {
    int en = e + stride;
    if (en < E) {
      int cn = cols[en];
      __builtin_prefetch(src + (size_t)cn * D + lane * 4, 0, 3);
    }
    float val = vals[e];
    int col = cols[e], row = rows[e];
    float4 v = ((const float4*)(src + (size_t)col * D))[lane];
    float* d = dst + (size_t)row * D + lane * 4;
    atomicAdd(d + 0, val * v.x);
    atomicAdd(d + 1, val * v.y);
    atomicAdd(d + 2, val * v.z);
    atomicAdd(d + 3, val * v.w);
  }
}

__global__ void zero2_kernel(float4* __restrict__ a, float4* __restrict__ b, int n4) {
  int i = blockIdx.x * blockDim.x + threadIdx.x;
  if (i < n4) {
    float4 z = make_float4(0.0f, 0.0f, 0.0f, 0.0f);
    a[i] = z; b[i] = z;
  }
}

__global__ void acc_zero_kernel(float4* __restrict__ acc, const float4* __restrict__ src,
                                float4* __restrict__ zbuf, int n4) {
  int i = blockIdx.x * blockDim.x + threadIdx.x;
  if (i < n4) {
    float4 a = acc[i], s = src[i];
    a.x += s.x; a.y += s.y; a.z += s.z; a.w += s.w;
    acc[i] = a;
    if (zbuf) zbuf[i] = make_float4(0.0f, 0.0f, 0.0f, 0.0f);
  }
}

// ---------------------------------------------------------------------------
// Final: y = proj(expmap0(acc, c=1), c=1), in place on d_out.
// ---------------------------------------------------------------------------
__global__ void expmap_kernel(float* __restrict__ io, int N) {
  int wave = (blockIdx.x * blockDim.x + threadIdx.x) >> 5;
  int lane = threadIdx.x & 31;
  if (wave >= N) return;

  float* row = io + (size_t)wave * D;
  float2 v = ((const float2*)row)[lane];
  float a = (lane == 0) ? 0.0f : v.x;   // u[:,1:] only
  float b = v.y;
  float s = a * a + b * b;
  #pragma unroll
  for (int m = 16; m >= 1; m >>= 1) s += __shfl_xor(s, m, 32);

  float xsq   = s;
  float xnorm = fmaxf(sqrtf(xsq), MIN_NORMF);
  float theta = fminf(xnorm, 15.0f);
  float et    = expf(theta);
  float sh    = 0.5f * (et - 1.0f / et);          // sinh
  float scale = sh / xnorm;
  float ysq   = scale * scale * xsq;              // |xr|^2 for re-projection
  float out0  = sqrtf(fmaxf(1.0f + ysq, EPSF));

  float2 o;
  o.x = (lane == 0) ? out0 : scale * v.x;
  o.y = scale * v.y;
  ((float2*)row)[lane] = o;
}

// ---------------------------------------------------------------------------
// Host-side orchestration (graph-capture safe: only launches on stream).
// ---------------------------------------------------------------------------
extern "C" void kernel_launch(void* const* d_in, const int* in_sizes, int n_in,
                              void* d_out, int out_size, void* d_ws, size_t ws_size,
                              hipStream_t stream) {
  const float* weight = (const float*)d_in[0];
  const float* vals   = (const float*)d_in[1];
  const int*   rows   = (const int*)  d_in[2];
  const int*   cols   = (const int*)  d_in[3];

  const int N  = in_sizes[0] / D;
  const int E  = in_sizes[1];
  const int nd = N * D;
  const int n4 = nd / 4;
  const int nb = (N + 255) / 256;     // scan blocks

  float* acc = (float*)d_out;
  float* A   = (float*)d_ws;
  float* B   = A + (size_t)nd;

  size_t need_csr = (size_t)2 * nd * 4        // A, B
                  + (size_t)2 * E * 4         // cval, cidx
                  + (size_t)(3 * N + nb) * 4; // counts, rowStart, cursor, blockSums

  // 1) A = x_tangent
  logmap_kernel<<<(N + 7) / 8, 256, 0, stream>>>(weight, A, N);

  if (ws_size >= need_csr) {
    // ---- CSR path: no float atomics in the hot loops ----
    float*    cval      = B + (size_t)nd;
    int*      cidx      = (int*)(cval + (size_t)E);
    unsigned* counts    = (unsigned*)(cidx + (size_t)E);
    unsigned* rowStart  = counts + N;
    unsigned* cursor    = rowStart + N;
    unsigned* blockSums = cursor + N;

    init_kernel<<<(n4 + 255) / 256, 256, 0, stream>>>((float4*)acc, counts, cursor, n4, N);
    count_kernel<<<(E + 255) / 256, 256, 0, stream>>>(rows, counts, E);
    scan_block_kernel<<<nb, 256, 0, stream>>>(counts, rowStart, blockSums, N);
    scan_top_kernel<<<1, 1, 0, stream>>>(blockSums, nb);
    scatter_kernel<<<(E + 255) / 256, 256, 0, stream>>>(vals, rows, cols, rowStart,
                                                        blockSums, cursor, cidx, cval, E);
    float* cur = A;
    float* nxt = B;
    for (int layer = 0; layer < 3; ++layer) {
      csr_spmm_kernel<<<(N + 15) / 16, 256, 0, stream>>>(cidx, cval, counts, rowStart,
                                                         blockSums, cur, nxt, acc, N);
      float* t = cur; cur = nxt; nxt = t;
    }
  } else {
    // ---- Fallback: COO scatter with f32 atomics (needs only A,B) ----
    zero2_kernel<<<(n4 + 255) / 256, 256, 0, stream>>>((float4*)acc, (float4*)B, n4);
    const int GPB = 16, ITERS = 4;
    int blocks = (E + GPB * ITERS - 1) / (GPB * ITERS);
    int stride = blocks * GPB;
    float* cur = A;
    float* nxt = B;
    for (int layer = 0; layer < 3; ++layer) {
      coo_spmm_kernel<<<blocks, 256, 0, stream>>>(vals, rows, cols, cur, nxt, E, stride);
      float4* zb = (layer < 2) ? (float4*)cur : (float4*)nullptr;
      acc_zero_kernel<<<(n4 + 255) / 256, 256, 0, stream>>>((float4*)acc,
                                                            (const float4*)nxt, zb, n4);
      float* t = cur; cur = nxt; nxt = t;
    }
  }

  // 4) d_out = proj(expmap0(acc)) in place
  expmap_kernel<<<(N + 7) / 8, 256, 0, stream>>>(acc, N);
}